// IdempotentCircorr2d_21431886807503
// MI455X (gfx1250) — compile-verified
//
#include <hip/hip_runtime.h>
#include <math.h>

// IdempotentCircorr2d on MI455X (gfx1250, wave32).
// Pipeline: FFT2(x) -> per-frequency {k_f, A=K K^H, complex Jacobi eig, top-16
// projector P, dist, y_f = conj(P) X via V_WMMA_F32_16X16X4_F32} -> IFFT2 + bias.
// The x_f tile is brought into LDS with GLOBAL_LOAD_ASYNC_TO_LDS_B128 and
// overlapped with the whole eigensolve (ASYNCcnt-tracked, gfx1250 async path).

#define PI_F 3.14159265358979323846f

typedef __attribute__((ext_vector_type(2))) float v2f;
typedef __attribute__((ext_vector_type(8))) float v8f;

__device__ __forceinline__ float2 cmulf(float2 a, float2 b) {
  return make_float2(a.x * b.x - a.y * b.y, a.x * b.y + a.y * b.x);
}

// 128-point iterative radix-2 FFT in LDS. Caller fills buf[0..127] and
// __syncthreads() before calling. Result is left in tmp[0..127].
// sign = -1 forward, +1 inverse (unnormalized sums).
__device__ void fft128_lds(const float2* buf, float2* tmp, int t, float sign) {
  unsigned r = __brev((unsigned)t) >> 25;  // 7-bit bit reversal
  tmp[t] = buf[r];
  __syncthreads();
  for (int len = 2; len <= 128; len <<= 1) {
    int half = len >> 1;
    if (t < 64) {
      int j = t & (half - 1);
      int i0 = (t / half) * len + j;
      int i1 = i0 + half;
      float ang = sign * 2.0f * PI_F * (float)j / (float)len;
      float sn, cs;
      __sincosf(ang, &sn, &cs);
      float2 w = make_float2(cs, sn);
      float2 a = tmp[i0];
      float2 b = cmulf(w, tmp[i1]);
      tmp[i0] = make_float2(a.x + b.x, a.y + b.y);
      tmp[i1] = make_float2(a.x - b.x, a.y - b.y);
    }
    __syncthreads();
  }
}

// ---------------- Pass F1: FFT along w. x (b,c,h,w) real -> X1 (b,c,h,w) cplx.
__global__ __launch_bounds__(128) void k_fft_rows(const float* __restrict__ x,
                                                  float2* __restrict__ X1) {
  __shared__ float2 s0[128], s1[128];
  int t = threadIdx.x;
  size_t row = blockIdx.x;  // (b*32+c)*128 + h, 65536 rows
  s0[t] = make_float2(x[row * 128 + t], 0.f);
  __syncthreads();
  fft128_lds(s0, s1, t, -1.f);
  X1[row * 128 + t] = s1[t];
}

// ------- Pass F2: FFT along h, apply ortho scale 1/128, transpose to (h,w,b,c).
__global__ __launch_bounds__(128) void k_fft_cols(const float2* __restrict__ X1,
                                                  float2* __restrict__ XF) {
  __shared__ float2 s0[128], s1[128];
  int t = threadIdx.x;
  int lin = blockIdx.x;  // (b*32+c)*128 + w
  int w = lin & 127;
  int bc = lin >> 7;
  int b = bc >> 5, c = bc & 31;
  s0[t] = X1[((size_t)bc * 128 + t) * 128 + w];  // gather column, h = t (L2-resident)
  __syncthreads();
  fft128_lds(s0, s1, t, -1.f);
  float2 v = s1[t];
  v.x *= (1.f / 128.f);
  v.y *= (1.f / 128.f);
  XF[(((size_t)t * 128 + w) * 16 + b) * 32 + c] = v;  // h = t
}

// ---------------- Per-frequency kernel: projector + WMMA GEMM ----------------
__global__ __launch_bounds__(128) void k_freq(const float* __restrict__ ker,   // (32,32,3,3)
                                              const float* __restrict__ bias, // (32,)
                                              float2* __restrict__ XF,        // in: x_f, out: y_f
                                              float* __restrict__ distAcc,
                                              float* __restrict__ biasNP) {
  __shared__ float2 Ks[1024];  // k_f (o,i)
  __shared__ float2 As[1024];  // A = K K^H (Hermitian), diagonalized in place
  __shared__ float2 Vs[1024];  // eigenvectors (columns)
  __shared__ float2 Ps[1024];  // projector
  __shared__ float2 Xs[512];   // x_f tile, b-major: Xs[b*32 + i]
  __shared__ float rcos_s[16];
  __shared__ float2 rs_s[16];
  __shared__ int pp_s[16], qq_s[16];
  __shared__ float sel_s[32];
  __shared__ float red[128];

  int t = threadIdx.x;
  int freq = blockIdx.x;  // 16384
  int fh = freq >> 7, fw = freq & 127;
  float2* xfreq = XF + (size_t)freq * 512;

  // Kick off the async copy of the 4KB x_f tile into LDS (layout-preserving,
  // b-major). Runs on the async engine, overlapped with the entire eigensolve;
  // consumed only by the WMMA section after s_wait_asynccnt + barrier.
  {
    unsigned xs_base = (unsigned)(uintptr_t)(&Xs[0]);  // low 32 bits = LDS offset
    const void* srcp = (const void*)xfreq;
#pragma unroll
    for (int it = 0; it < 2; ++it) {
      int u = t + it * 128;  // 256 chunks of 16B = 4KB
      asm volatile("global_load_async_to_lds_b128 %0, %1, %2"
                   :
                   : "v"(xs_base + u * 16), "v"(u * 16), "s"(srcp)
                   : "memory");
    }
  }

  // k_f[o][i] = sum_{p,q<3} ker[o][i][p][q] * e^{-2pi i (fh*p + fw*q)/128}.
  float twr[9], twi[9];
  for (int p = 0; p < 3; ++p)
    for (int q = 0; q < 3; ++q) {
      float ang = -2.f * PI_F * (float)(fh * p + fw * q) * (1.f / 128.f);
      __sincosf(ang, &twi[p * 3 + q], &twr[p * 3 + q]);
    }
  for (int u = t; u < 1024; u += 128) {
    const float* kp = ker + u * 9;  // u = o*32 + i
    float sr = 0.f, si = 0.f;
#pragma unroll
    for (int e = 0; e < 9; ++e) {
      sr += kp[e] * twr[e];
      si += kp[e] * twi[e];
    }
    Ks[u] = make_float2(sr, si);
  }
  __syncthreads();

  // A = K K^H ; V = I.
  for (int u = t; u < 1024; u += 128) {
    int r = u >> 5, cidx = u & 31;
    float ar = 0.f, ai = 0.f;
    for (int i = 0; i < 32; ++i) {
      float2 a = Ks[r * 32 + i], b = Ks[cidx * 32 + i];
      ar += a.x * b.x + a.y * b.y;   // a * conj(b)
      ai += a.y * b.x - a.x * b.y;
    }
    As[u] = make_float2(ar, ai);
    Vs[u] = make_float2((r == cidx) ? 1.f : 0.f, 0.f);
  }
  __syncthreads();

  // Cyclic complex Jacobi: 8 sweeps x 31 rounds, 16 disjoint pairs per round.
  for (int sweep = 0; sweep < 8; ++sweep) {
    for (int rnd = 0; rnd < 31; ++rnd) {
      if (t < 16) {
        int k = t;
        int p = (k == 0) ? 0 : ((k - 1 + rnd) % 31) + 1;  // circle method
        int q = ((30 - k + rnd) % 31) + 1;
        if (p > q) { int tm = p; p = q; q = tm; }
        float2 apq = As[p * 32 + q];
        float app = As[p * 32 + p].x;
        float aqq = As[q * 32 + q].x;
        float mag = sqrtf(apq.x * apq.x + apq.y * apq.y);
        float rc;
        float2 rs;
        if (mag < 1e-30f) {
          rc = 1.f;
          rs = make_float2(0.f, 0.f);
        } else {
          float tau = (aqq - app) / (2.f * mag);
          float tt = ((tau >= 0.f) ? 1.f : -1.f) / (fabsf(tau) + sqrtf(1.f + tau * tau));
          rc = rsqrtf(1.f + tt * tt);
          float srm = tt * rc;
          rs = make_float2(srm * apq.x / mag, srm * apq.y / mag);
        }
        rcos_s[k] = rc;
        rs_s[k] = rs;
        pp_s[k] = p;
        qq_s[k] = q;
      }
      __syncthreads();
      // Column updates: A <- A*J and V <- V*J  (1024 tasks).
      for (int u = t; u < 1024; u += 128) {
        float2* M = (u >> 9) ? Vs : As;
        int k = (u >> 5) & 15;
        int r = u & 31;
        int p = pp_s[k], q = qq_s[k];
        float rc = rcos_s[k];
        float2 s = rs_s[k];
        float2 uu = M[r * 32 + p], vv = M[r * 32 + q];
        // new_p = c*u - conj(s)*v ; new_q = s*u + c*v
        float2 np = make_float2(rc * uu.x - (s.x * vv.x + s.y * vv.y),
                                rc * uu.y - (s.x * vv.y - s.y * vv.x));
        float2 nq = make_float2(s.x * uu.x - s.y * uu.y + rc * vv.x,
                                s.x * uu.y + s.y * uu.x + rc * vv.y);
        M[r * 32 + p] = np;
        M[r * 32 + q] = nq;
      }
      __syncthreads();
      // Row updates: A <- J^H * A  (512 tasks).
      for (int u = t; u < 512; u += 128) {
        int k = u >> 5;
        int cc = u & 31;
        int p = pp_s[k], q = qq_s[k];
        float rc = rcos_s[k];
        float2 s = rs_s[k];
        float2 uu = As[p * 32 + cc], vv = As[q * 32 + cc];
        float2 sv = cmulf(s, vv);
        float2 np = make_float2(rc * uu.x - sv.x, rc * uu.y - sv.y);
        float2 cu = make_float2(s.x * uu.x + s.y * uu.y,   // conj(s)*u
                                s.x * uu.y - s.y * uu.x);
        float2 nq = make_float2(cu.x + rc * vv.x, cu.y + rc * vv.y);
        As[p * 32 + cc] = np;
        As[q * 32 + cc] = nq;
      }
      __syncthreads();
    }
  }

  // Select top-16 eigenvalues (rank with index tie-break).
  if (t < 32) {
    float lam = As[t * 32 + t].x;
    int rank = 0;
    for (int m = 0; m < 32; ++m) {
      float lm = As[m * 32 + m].x;
      if (lm > lam || (lm == lam && m < t)) ++rank;
    }
    sel_s[t] = (rank < 16) ? 1.f : 0.f;
  }
  __syncthreads();

  // P[r][c] = sum_{j in top16} V[r][j] * conj(V[c][j]).
  for (int u = t; u < 1024; u += 128) {
    int r = u >> 5, cc = u & 31;
    float pr = 0.f, pi = 0.f;
    for (int j = 0; j < 32; ++j) {
      float sj = sel_s[j];
      float2 a = Vs[r * 32 + j], b = Vs[cc * 32 + j];
      pr += sj * (a.x * b.x + a.y * b.y);
      pi += sj * (a.y * b.x - a.x * b.y);
    }
    Ps[u] = make_float2(pr, pi);
  }
  __syncthreads();

  // dist contribution: sum |k_f - P|^2.
  float dloc = 0.f;
  for (int u = t; u < 1024; u += 128) {
    float dx = Ks[u].x - Ps[u].x;
    float dy = Ks[u].y - Ps[u].y;
    dloc += dx * dx + dy * dy;
  }
  red[t] = dloc;
  __syncthreads();
  for (int off = 64; off > 0; off >>= 1) {
    if (t < off) red[t] += red[t + off];
    __syncthreads();
  }
  if (t == 0) atomicAdd(distAcc, red[0]);

  // Bias correction from the DC projector: bn = bias - Re(P) @ bias.
  if (freq == 0 && t < 32) {
    float acc = 0.f;
    for (int i = 0; i < 32; ++i) acc += Ps[t * 32 + i].x * bias[i];
    biasNP[t] = bias[t] - acc;
  }

  // Drain the async x_f tile copy (per-wave ASYNCcnt), then make it visible
  // to all waves before the WMMA section reads it from LDS.
  asm volatile("s_wait_asynccnt 0x0" ::: "memory");
  __syncthreads();

  // y_f[o][b] = sum_i conj(P)[o][i] * X[i][b], M=32 N=16 K=32 complex.
  // conj(P)*X: re = Pr*Xr + Pi*Xi ; im = Pr*Xi - Pi*Xr.
  // Four f32 WMMA accumulator chains (V_WMMA_F32_16X16X4_F32), 2 M-tiles.
  int wave = t >> 5;
  int lane = t & 31;
  if (wave < 2) {
    int mt = wave;                 // M-tile (rows 16*mt .. 16*mt+15)
    int m = mt * 16 + (lane & 15); // A-fragment row for this lane
    int nb = lane & 15;            // B-fragment column for this lane
    int kk = lane >> 4;            // K sub-group (K offset 2*kk)
    v8f crr = {}, cii = {}, cri = {}, cir = {};
#pragma unroll
    for (int k0 = 0; k0 < 32; k0 += 4) {
      v2f apr, api, bxr, bxi;
#pragma unroll
      for (int v = 0; v < 2; ++v) {
        int k = k0 + 2 * kk + v;
        float2 pv = Ps[m * 32 + k];
        apr[v] = pv.x;
        api[v] = pv.y;
        float2 xv = Xs[nb * 32 + k];  // b-major LDS tile
        bxr[v] = xv.x;
        bxi[v] = xv.y;
      }
      crr = __builtin_amdgcn_wmma_f32_16x16x4_f32(false, apr, false, bxr, (short)0, crr, false, false);
      cii = __builtin_amdgcn_wmma_f32_16x16x4_f32(false, api, false, bxi, (short)0, cii, false, false);
      cri = __builtin_amdgcn_wmma_f32_16x16x4_f32(false, apr, false, bxi, (short)0, cri, false, false);
      cir = __builtin_amdgcn_wmma_f32_16x16x4_f32(false, api, false, bxr, (short)0, cir, false, false);
    }
    // C/D layout: VGPR v holds row v (lanes 0-15) / row v+8 (lanes 16-31).
#pragma unroll
    for (int v = 0; v < 8; ++v) {
      int o = mt * 16 + v + ((lane >> 4) << 3);
      int b = lane & 15;
      float re = crr[v] + cii[v];
      float im = cri[v] - cir[v];
      xfreq[b * 32 + o] = make_float2(re, im);  // in-place y_f
    }
  }
}

// ---------------- Pass I1: IFFT along h. YF (h,w,b,c) -> Y1 (b,c,h,w).
__global__ __launch_bounds__(128) void k_ifft_cols(const float2* __restrict__ YF,
                                                   float2* __restrict__ Y1) {
  __shared__ float2 s0[128], s1[128];
  int t = threadIdx.x;
  int lin = blockIdx.x;  // w*512 + b*32 + c
  int c = lin & 31;
  int b = (lin >> 5) & 15;
  int w = lin >> 9;
  s0[t] = YF[(((size_t)t * 128 + w) * 16 + b) * 32 + c];  // h = t (L2-resident)
  __syncthreads();
  fft128_lds(s0, s1, t, +1.f);
  int bc = b * 32 + c;
  Y1[((size_t)bc * 128 + t) * 128 + w] = s1[t];
}

// ------- Pass I2: IFFT along w, ortho scale, real part, + bias projection.
__global__ __launch_bounds__(128) void k_ifft_rows(const float2* __restrict__ Y1,
                                                   const float* __restrict__ biasNP,
                                                   float* __restrict__ out) {
  __shared__ float2 s0[128], s1[128];
  int t = threadIdx.x;
  int lin = blockIdx.x;  // (b*32+c)*128 + h
  int c = (lin >> 7) & 31;
  s0[t] = Y1[(size_t)lin * 128 + t];
  __syncthreads();
  fft128_lds(s0, s1, t, +1.f);
  out[(size_t)lin * 128 + t] = s1[t].x * (1.f / 128.f) + biasNP[c];
}

__global__ void k_init(float* __restrict__ acc) {
  if (threadIdx.x == 0 && blockIdx.x == 0) acc[0] = 0.f;
}

__global__ void k_finish(const float* __restrict__ acc, float* __restrict__ out, int n) {
  if (threadIdx.x == 0 && blockIdx.x == 0) out[n] = sqrtf(acc[0]);
}

extern "C" void kernel_launch(void* const* d_in, const int* in_sizes, int n_in,
                              void* d_out, int out_size, void* d_ws, size_t ws_size,
                              hipStream_t stream) {
  const float* x = (const float*)d_in[0];     // (16,32,128,128)
  const float* ker = (const float*)d_in[1];   // (32,32,3,3)
  const float* bias = (const float*)d_in[2];  // (32,)
  float* out = (float*)d_out;                 // 16*32*128*128 floats + 1 (dist)
  char* ws = (char*)d_ws;

  const size_t NC = 16ull * 32 * 128 * 128;  // 8,388,608 complex slots
  float2* X1 = (float2*)ws;                  // 64 MB: (b,c,h,w) spectral temp
  float2* XF = (float2*)(ws + NC * sizeof(float2));  // 64 MB: (h,w,b,c) x_f / y_f
  float* acc = (float*)(ws + 2 * NC * sizeof(float2));
  float* biasNP = (float*)(ws + 2 * NC * sizeof(float2) + 64);

  k_init<<<1, 32, 0, stream>>>(acc);
  k_fft_rows<<<16 * 32 * 128, 128, 0, stream>>>(x, X1);
  k_fft_cols<<<16 * 32 * 128, 128, 0, stream>>>(X1, XF);
  k_freq<<<128 * 128, 128, 0, stream>>>(ker, bias, XF, acc, biasNP);
  k_ifft_cols<<<128 * 16 * 32, 128, 0, stream>>>(XF, X1);
  k_ifft_rows<<<16 * 32 * 128, 128, 0, stream>>>(X1, biasNP, out);
  k_finish<<<1, 32, 0, stream>>>(acc, out, (int)NC);
}